// LSTMTransformerModel_44126493999777
// MI455X (gfx1250) — compile-verified
//
#include <hip/hip_runtime.h>
#include <math.h>

typedef __attribute__((ext_vector_type(2))) float v2f;
typedef __attribute__((ext_vector_type(8))) float v8f;

__device__ __forceinline__ float sigm(float v) { return 1.0f / (1.0f + expf(-v)); }

// ---------------------------------------------------------------------------
// Kernel 0: _prepare — tiny scalar geometry prep, one thread.
// Produces xs (5 x 42) = flattened [leftN, rightN, f2l, f2r, h2h] (105 x 2).
// ---------------------------------------------------------------------------
__global__ void prep_kern(const float* __restrict__ p, float* __restrict__ xs) {
    if (threadIdx.x != 0 || blockIdx.x != 0) return;
    const float sbx = p[0],   sby = p[1];     // src_body  = p[0]
    const float slx = p[200], sly = p[201];   // src_left  = p[100]
    const float srx = p[242], sry = p[243];   // src_right = p[121]
    const float* L = p + 182;                 // left  = p[91:112]
    const float* R = p + 224;                 // right = p[112:133]
    float lminx = L[0], lmaxx = L[0], lminy = L[1], lmaxy = L[1];
    float rminx = R[0], rmaxx = R[0], rminy = R[1], rmaxy = R[1];
    for (int i = 1; i < 21; ++i) {
        float a = L[2*i], b = L[2*i+1];
        lminx = fminf(lminx, a); lmaxx = fmaxf(lmaxx, a);
        lminy = fminf(lminy, b); lmaxy = fmaxf(lmaxy, b);
        a = R[2*i]; b = R[2*i+1];
        rminx = fminf(rminx, a); rmaxx = fmaxf(rmaxx, a);
        rminy = fminf(rminy, b); rmaxy = fmaxf(rmaxy, b);
    }
    const float wl = lmaxx - lminx, hl = lmaxy - lminy;
    const float wr = rmaxx - rminx, hr = rmaxy - rminy;
    const float wb = p[10] - p[12];           // body[5,0] - body[6,0]
    const float hb4 = 4.0f * wb;
    const bool  cl = (wl != 0.0f) && (hl != 0.0f);
    const bool  cr = (wr != 0.0f) && (hr != 0.0f);
    const bool  cb = (wb != 0.0f) && (hb4 != 0.0f);
    const float dwl = cl ? wl : 1.0f, dhl = cl ? hl : 1.0f;
    const float dwr = cr ? wr : 1.0f, dhr = cr ? hr : 1.0f;
    const float dwb = cb ? wb : 1.0f, dhb = cb ? hb4 : 1.0f;
    for (int i = 0; i < 21; ++i) {
        const float lx = L[2*i], ly = L[2*i+1];
        const float rx = R[2*i], ry = R[2*i+1];
        xs[      2*i] = (lx - slx) / dwl;  xs[      2*i+1] = (ly - sly) / dhl;
        xs[ 42 + 2*i] = (rx - srx) / dwr;  xs[ 42 + 2*i+1] = (ry - sry) / dhr;
        xs[ 84 + 2*i] = (lx - sbx) / dwb;  xs[ 84 + 2*i+1] = (ly - sby);
        xs[126 + 2*i] = (rx - sbx);        xs[126 + 2*i+1] = (ry - sby) / dhb;
        xs[168 + 2*i] = (lx - rx)  / dwb;  xs[168 + 2*i+1] = (ly - ry)  / dhb;
    }
}

// ---------------------------------------------------------------------------
// Kernel 1: layer 0 (K=42, tiny) — thread per (s,j); also zero-init gbuf.
// Only gates i (row j), g (row j+2048), o (row j+3072) are live (c0=h0=0).
// ---------------------------------------------------------------------------
__global__ __launch_bounds__(256) void layer0_kern(
        const float* __restrict__ W_ih0, const float* __restrict__ b_ih,
        const float* __restrict__ b_hh, const float* __restrict__ xsbuf,
        float* __restrict__ inp0, float* __restrict__ gbuf) {
    __shared__ float xv[42];
    const int tid = blockIdx.x * 256 + threadIdx.x;   // 0..5119
    const int s = tid >> 10, j = tid & 1023;          // s uniform per block
    if (threadIdx.x < 42) xv[threadIdx.x] = xsbuf[s * 42 + threadIdx.x];
    __syncthreads();
    const int goff[3] = {0, 2, 3};
    float g3[3];
    for (int t = 0; t < 3; ++t) {
        const int row = goff[t] * 1024 + j;
        const float* wp = W_ih0 + ((size_t)s * 4096 + row) * 42;
        float sum = b_ih[(s * 4 + 0) * 4096 + row] + b_hh[(s * 4 + 0) * 4096 + row];
        #pragma unroll
        for (int k = 0; k < 42; ++k) sum += wp[k] * xv[k];
        g3[t] = sum;
    }
    const float cc = sigm(g3[0]) * tanhf(g3[1]);
    inp0[tid] = sigm(g3[2]) * tanhf(cc);
    gbuf[(s * 3 + 0) * 1024 + j] = 0.0f;
    gbuf[(s * 3 + 1) * 1024 + j] = 0.0f;
    gbuf[(s * 3 + 2) * 1024 + j] = 0.0f;
}

// ---------------------------------------------------------------------------
// Kernel 2: layers 1..3 gate GEMV via V_WMMA_F32_16X16X4_F32.
// One wave = 16 rows of one gate, one K-chunk of 128. 7680 waves = 960 blocks
// (~7.5 waves/SIMD on a 1024-SIMD part for HBM latency hiding).
// A: lanes 0-15 hold rows m=0..15 k,k+1; lanes 16-31 hold k+2,k+3 (ISA layout).
// B: x broadcast to all 16 columns -> D columns all equal y; lane 0 has
// y0..7 in c[0..7], lane 16 has y8..15. Partials combined with f32 atomics.
// ---------------------------------------------------------------------------
__global__ __launch_bounds__(256) void gate_gemv(
        const float* __restrict__ Wihr, const float* __restrict__ xin,
        float* __restrict__ gbuf, int lm1) {
    __shared__ float xsh[1024];
    const int lane = threadIdx.x & 31;
    const int u = blockIdx.x * 8 + (threadIdx.x >> 5);  // 0..7679
    const int kc = u & 7;                               // K chunk (128)
    const int jg = (u >> 3) & 63;                       // j group of 16
    const int t  = (u >> 9) % 3;                        // gate: i,g,o
    const int s  = u / 1536;                            // uniform per block
    // stage x[s] (4 KB) coalesced
    ((float4*)xsh)[threadIdx.x] = ((const float4*)(xin + s * 1024))[threadIdx.x];
    __syncthreads();
    const int goff[3] = {0, 2, 3};
    const int off = (lane >> 4) << 1;                   // 0 or 2 (K half-split)
    const float* wp = Wihr
        + ((size_t)(s * 3 + lm1) * 4096 + goff[t] * 1024 + jg * 16 + (lane & 15)) * 1024
        + kc * 128 + off;
    const float* xp = xsh + kc * 128 + off;
    v8f c = {0.f,0.f,0.f,0.f,0.f,0.f,0.f,0.f};
    #pragma unroll 4
    for (int kk = 0; kk < 128; kk += 4) {
        v2f a = *(const v2f*)(wp + kk);
        v2f b = *(const v2f*)(xp + kk);
        c = __builtin_amdgcn_wmma_f32_16x16x4_f32(false, a, false, b,
                                                  (short)0, c, false, false);
    }
    float* gout = gbuf + (s * 3 + t) * 1024 + jg * 16;
    if (lane == 0) {
        #pragma unroll
        for (int r = 0; r < 8; ++r) unsafeAtomicAdd(gout + r, c[r]);
    } else if (lane == 16) {
        #pragma unroll
        for (int r = 0; r < 8; ++r) unsafeAtomicAdd(gout + 8 + r, c[r]);
    }
}

// ---------------------------------------------------------------------------
// Kernel 3: activation — thread per (s,j): add biases, LSTM nonlinearity,
// re-zero gbuf for the next layer; layer 3 also zeroes fbuf for fuse stage.
// ---------------------------------------------------------------------------
__global__ __launch_bounds__(256) void act_kern(
        const float* __restrict__ b_ih, const float* __restrict__ b_hh,
        float* __restrict__ gbuf, float* __restrict__ outv,
        float* __restrict__ fbuf, int layer, int zero_fbuf) {
    const int tid = blockIdx.x * 256 + threadIdx.x;   // 0..5119
    const int s = tid >> 10, j = tid & 1023;
    float gi = gbuf[(s * 3 + 0) * 1024 + j];
    float gg = gbuf[(s * 3 + 1) * 1024 + j];
    float go = gbuf[(s * 3 + 2) * 1024 + j];
    gbuf[(s * 3 + 0) * 1024 + j] = 0.0f;
    gbuf[(s * 3 + 1) * 1024 + j] = 0.0f;
    gbuf[(s * 3 + 2) * 1024 + j] = 0.0f;
    const size_t bb = (size_t)(s * 4 + layer) * 4096;
    gi += b_ih[bb + j]        + b_hh[bb + j];
    gg += b_ih[bb + 2048 + j] + b_hh[bb + 2048 + j];
    go += b_ih[bb + 3072 + j] + b_hh[bb + 3072 + j];
    const float cc = sigm(gi) * tanhf(gg);
    outv[tid] = sigm(go) * tanhf(cc);
    if (zero_fbuf && tid < 1024) fbuf[tid] = 0.0f;
}

// ---------------------------------------------------------------------------
// Kernel 4: fuse GEMV (1024 x 5120) via WMMA, K chunks of 256 (1280 waves).
// ---------------------------------------------------------------------------
__global__ __launch_bounds__(256) void fuse_gemv(
        const float* __restrict__ fuse_w, const float* __restrict__ xin,
        float* __restrict__ fbuf) {
    __shared__ float xsh[5120];
    for (int i = threadIdx.x; i < 5120; i += 256) xsh[i] = xin[i];
    __syncthreads();
    const int lane = threadIdx.x & 31;
    const int u = blockIdx.x * 8 + (threadIdx.x >> 5);  // 0..1279
    const int kc = u % 20;
    const int jg = u / 20;
    const int off = (lane >> 4) << 1;
    const float* wp = fuse_w + ((size_t)(jg * 16 + (lane & 15))) * 5120 + kc * 256 + off;
    const float* xp = xsh + kc * 256 + off;
    v8f c = {0.f,0.f,0.f,0.f,0.f,0.f,0.f,0.f};
    #pragma unroll 4
    for (int kk = 0; kk < 256; kk += 4) {
        v2f a = *(const v2f*)(wp + kk);
        v2f b = *(const v2f*)(xp + kk);
        c = __builtin_amdgcn_wmma_f32_16x16x4_f32(false, a, false, b,
                                                  (short)0, c, false, false);
    }
    float* gout = fbuf + jg * 16;
    if (lane == 0) {
        #pragma unroll
        for (int r = 0; r < 8; ++r) unsafeAtomicAdd(gout + r, c[r]);
    } else if (lane == 16) {
        #pragma unroll
        for (int r = 0; r < 8; ++r) unsafeAtomicAdd(gout + 8 + r, c[r]);
    }
}

// ---------------------------------------------------------------------------
// Kernel 5: fc — ctx = fbuf + fuse_b (softmax over len-1 axis == identity),
// out = fc_w @ ctx + fc_b. Wave per row, shuffle reduce. 2 MB, trivial.
// ---------------------------------------------------------------------------
__global__ __launch_bounds__(256) void fc_kern(
        const float* __restrict__ fc_w, const float* __restrict__ fc_b,
        const float* __restrict__ fuse_b, const float* __restrict__ fbuf,
        float* __restrict__ out) {
    __shared__ float ctx[1024];
    for (int i = threadIdx.x; i < 1024; i += 256) ctx[i] = fbuf[i] + fuse_b[i];
    __syncthreads();
    const int r = blockIdx.x * 8 + (threadIdx.x >> 5);
    if (r >= 500) return;
    const int lane = threadIdx.x & 31;
    const float* wp = fc_w + (size_t)r * 1024;
    float sum = 0.0f;
    #pragma unroll
    for (int k = lane * 4; k < 1024; k += 128) {
        const float4 w4 = *(const float4*)(wp + k);
        sum += w4.x * ctx[k] + w4.y * ctx[k+1] + w4.z * ctx[k+2] + w4.w * ctx[k+3];
    }
    for (int d = 16; d > 0; d >>= 1) sum += __shfl_down(sum, d, 32);
    if (lane == 0) out[r] = sum + fc_b[r];
}

// ---------------------------------------------------------------------------
extern "C" void kernel_launch(void* const* d_in, const int* in_sizes, int n_in,
                              void* d_out, int out_size, void* d_ws, size_t ws_size,
                              hipStream_t stream) {
    const float* x      = (const float*)d_in[0];
    const float* W_ih0  = (const float*)d_in[1];
    const float* W_ihr  = (const float*)d_in[2];
    // d_in[3] = W_hh: multiplies h0 == 0, never read.
    const float* b_ih   = (const float*)d_in[4];
    const float* b_hh   = (const float*)d_in[5];
    const float* fuse_w = (const float*)d_in[6];
    const float* fuse_b = (const float*)d_in[7];
    // d_in[8..11] = att_w/att_b/score_w/score_b: softmax over length-1 axis
    // is exactly 1.0, so attention is identity; never read.
    const float* fc_w   = (const float*)d_in[12];
    const float* fc_b   = (const float*)d_in[13];
    float* out = (float*)d_out;

    float* ws   = (float*)d_ws;
    float* XS   = ws;                // 210  (pad 256)
    float* INPA = ws + 256;          // 5120
    float* INPB = ws + 256 + 5120;   // 5120
    float* GBUF = ws + 256 + 10240;  // 15360 (s x {i,g,o} x 1024)
    float* FBUF = GBUF + 15360;      // 1024

    prep_kern<<<1, 32, 0, stream>>>(x, XS);
    layer0_kern<<<20, 256, 0, stream>>>(W_ih0, b_ih, b_hh, XS, INPA, GBUF);

    float* bufs[2] = {INPA, INPB};
    for (int layer = 1; layer <= 3; ++layer) {
        float* cur = bufs[(layer + 1) & 1];
        float* nxt = bufs[layer & 1];
        gate_gemv<<<960, 256, 0, stream>>>(W_ihr, cur, GBUF, layer - 1);
        act_kern<<<20, 256, 0, stream>>>(b_ih, b_hh, GBUF, nxt, FBUF, layer,
                                         layer == 3 ? 1 : 0);
    }
    fuse_gemv<<<160, 256, 0, stream>>>(fuse_w, INPB, FBUF);
    fc_kern<<<63, 256, 0, stream>>>(fc_w, fc_b, fuse_b, FBUF, out);
}